// FactorizedAntisymmetrize_60765197304505
// MI455X (gfx1250) — compile-verified
//
#include <hip/hip_runtime.h>
#include <stdint.h>
#include <math.h>

// ---------------------------------------------------------------------------
// FactorizedAntisymmetrize on gfx1250 (MI455X, wave32)
//
// psi[b] = sum_p sign(p) * ( V . tanh( W^T xp + b ) ),  xp = perm-gathered x
// Key identity: (xp @ W)[h] = sum_s  x[perm[s]] . W[s*16:(s+1)*16, h]
// => precompute P[b,i,s,h] with WMMA (fp32, exact), then per-permutation work
//    is 7 LDS gathers + tanh + dot.  ~20x fewer FLOPs than the naive GEMM.
// Hot loop tanh uses the CDNA5 hardware transcendental v_tanh_f32.
// ---------------------------------------------------------------------------

typedef __attribute__((ext_vector_type(2))) float v2f;
typedef __attribute__((ext_vector_type(8))) float v8f;

#define NPERM   5040
#define B_SZ    128
#define NPART   7
#define DDIM    16
#define HDIM    128
#define CHUNKS  4
#define PER_CHUNK (NPERM / CHUNKS)          // 1260
#define P_SLAB  (NPART * NPART * HDIM)      // 6272 floats per (b, leaf)

// workspace layout (float indices)
#define WS_CODES 0                           // 5040 x u32
#define WS_P0    8192                        // 128 * 6272 floats
#define WS_P1    (WS_P0 + B_SZ * P_SLAB)
#define WS_PART  (WS_P1 + B_SZ * P_SLAB)     // 2 * 128 * CHUNKS floats
// total floats = WS_PART + 1024  (~6.46 MB)

// ---------------------------------------------------------------------------
// Kernel 0: packed lexicographic permutation codes + parity sign.
// p -> Lehmer digits d_k (mixed factorial radix); perm value packed 3b/slot,
// sign(p) = (-1)^{sum d_k} stored in bit 31.
// ---------------------------------------------------------------------------
__global__ void build_perm_codes(uint32_t* __restrict__ codes) {
  int p = blockIdx.x * blockDim.x + threadIdx.x;
  if (p >= NPERM) return;
  const int fact[7] = {720, 120, 24, 6, 2, 1, 1};
  int lst[7] = {0, 1, 2, 3, 4, 5, 6};
  int rem = p, par = 0;
  uint32_t code = 0;
  for (int pos = 0; pos < 7; ++pos) {
    int d = rem / fact[pos];
    rem -= d * fact[pos];
    par += d;
    int v = lst[d];
    for (int j = d; j < 6; ++j) lst[j] = lst[j + 1];   // remove chosen element
    code |= (uint32_t)v << (3 * pos);
  }
  code |= ((uint32_t)(par & 1)) << 31;
  codes[p] = code;
}

// ---------------------------------------------------------------------------
// Kernel 1: P[b,i,s,h] = sum_k x[b,i,k] * W[s*16+k, h]  via WMMA fp32.
// A = x viewed as (896 x 16) row-major; per s: B_s = W[s*16 .. s*16+15, 0:128].
// Grid: (Mtile=56, s=7, leaf=2), 256 threads = 8 waves, wave w -> N-tile w.
// Four K=4 WMMA steps accumulate the K=16 contraction.
// VGPR layouts per CDNA5 ISA 7.12.2:
//   A 16x4 f32 : lane m(0..15)/half: vgpr0 = A[m][2*half], vgpr1 = A[m][2*half+1]
//   B 4x16 f32 : lane n(0..15)/half: vgpr0 = B[2*half][n], vgpr1 = B[2*half+1][n]
//   D 16x16 f32: vgpr r: lanes0-15 M=r, lanes16-31 M=8+r, N=lane%16
// ---------------------------------------------------------------------------
__global__ void precompute_P_wmma(const float* __restrict__ x0, const float* __restrict__ W0,
                                  const float* __restrict__ x1, const float* __restrict__ W1,
                                  float* __restrict__ ws) {
  const int mtile = blockIdx.x;        // 0..55  (56*16 == 128*7 rows exactly)
  const int s     = blockIdx.y;        // 0..6
  const int leaf  = blockIdx.z;        // 0..1
  const int lane  = threadIdx.x & 31;
  const int wtile = threadIdx.x >> 5;  // 0..7  -> h columns [16*wtile, 16*wtile+16)

  const float* x = leaf ? x1 : x0;
  const float* W = leaf ? W1 : W0;
  float* P = ws + (leaf ? WS_P1 : WS_P0);

  const int m    = lane & 15;
  const int half = lane >> 4;
  const float* arow = x + (mtile * 16 + m) * DDIM;   // contiguous K
  const int col = wtile * 16 + m;                    // N index for B/D

  v8f acc = {0.f, 0.f, 0.f, 0.f, 0.f, 0.f, 0.f, 0.f};
#pragma unroll
  for (int kf = 0; kf < 4; ++kf) {
    const int kk = kf * 4 + 2 * half;
    v2f a, b;
    a.x = arow[kk];
    a.y = arow[kk + 1];
    b.x = W[(s * 16 + kk) * HDIM + col];
    b.y = W[(s * 16 + kk + 1) * HDIM + col];
    acc = __builtin_amdgcn_wmma_f32_16x16x4_f32(
        /*neg_a=*/false, a, /*neg_b=*/false, b,
        /*c_mod=*/(short)0, acc, /*reuse_a=*/false, /*reuse_b=*/false);
  }

#pragma unroll
  for (int r = 0; r < 8; ++r) {
    int row = mtile * 16 + 8 * half + r;             // row = b*7 + i
    P[(row * NPART + s) * HDIM + wtile * 16 + m] = acc[r];
  }
}

// ---------------------------------------------------------------------------
// Kernel 2: per (chunk, b, leaf): partial psi = sum_p sign(p)*(V.tanh(bias+sum_s P[perm_s,s,:]))
// P[b] slab (25 KB) + code chunk staged in LDS; lane owns h = 4*lane..4*lane+3
// (ds_load_b128, lane-contiguous => conflict-free); deterministic reductions.
// tanh via CDNA5 hardware v_tanh_f32 (TRANS32 pipe, co-executes with VALU).
// ISA §7.4: TRANS result needs 1 independent op / V_NOP before use -> the
// four chained tanhs separate each other; trailing v_nop covers the last.
// ---------------------------------------------------------------------------
__global__ void __launch_bounds__(256)
perm_sum_kernel(const float* __restrict__ bias0, const float* __restrict__ V0,
                const float* __restrict__ bias1, const float* __restrict__ V1,
                float* __restrict__ ws) {
  const int chunk = blockIdx.x;   // 0..CHUNKS-1
  const int b     = blockIdx.y;   // 0..127
  const int leaf  = blockIdx.z;   // 0..1
  const int tid   = threadIdx.x;
  const int lane  = tid & 31;
  const int wave  = tid >> 5;     // 0..7

  __shared__ float    sP[P_SLAB];         // 25088 B
  __shared__ uint32_t sCodes[PER_CHUNK];  //  5040 B
  __shared__ float    sRed[8];

  const float* P = ws + (leaf ? WS_P1 : WS_P0) + b * P_SLAB;
  const uint32_t* gcodes = (const uint32_t*)ws;

  const float4* gP4 = (const float4*)P;
  float4* sP4 = (float4*)sP;
  for (int t = tid; t < P_SLAB / 4; t += blockDim.x) sP4[t] = gP4[t];
  for (int t = tid; t < PER_CHUNK; t += blockDim.x)
    sCodes[t] = gcodes[chunk * PER_CHUNK + t];
  __syncthreads();

  const float* bias = leaf ? bias1 : bias0;
  const float* V    = leaf ? V1 : V0;
  const float4 v4 = *(const float4*)(V + 4 * lane);
  const float4 bb = *(const float4*)(bias + 4 * lane);

  float acc = 0.f;
  for (int pl = wave; pl < PER_CHUNK; pl += 8) {
    uint32_t code = sCodes[pl];
    float4 sum = bb;
#pragma unroll
    for (int s = 0; s < NPART; ++s) {
      int idx = (code >> (3 * s)) & 7;
      float4 pv = sP4[(idx * NPART + s) * 32 + lane];   // ds_load_b128
      sum.x += pv.x; sum.y += pv.y; sum.z += pv.z; sum.w += pv.w;
    }
    float t0, t1, t2, t3;
    asm("v_tanh_f32 %0, %4\n\t"
        "v_tanh_f32 %1, %5\n\t"
        "v_tanh_f32 %2, %6\n\t"
        "v_tanh_f32 %3, %7\n\t"
        "v_nop"
        : "=&v"(t0), "=&v"(t1), "=&v"(t2), "=v"(t3)
        : "v"(sum.x), "v"(sum.y), "v"(sum.z), "v"(sum.w));
    float dot = t0 * v4.x + t1 * v4.y + t2 * v4.z + t3 * v4.w;
    // permutation sign lives in bit 31 of code: flip sign bit of dot directly
    uint32_t di = __float_as_uint(dot) ^ (code & 0x80000000u);
    acc += __uint_as_float(di);
  }

  // wave32 reduction
#pragma unroll
  for (int off = 16; off > 0; off >>= 1)
    acc += __shfl_down(acc, off, 32);
  if (lane == 0) sRed[wave] = acc;
  __syncthreads();
  if (tid == 0) {
    float s = 0.f;
#pragma unroll
    for (int wv = 0; wv < 8; ++wv) s += sRed[wv];
    ws[WS_PART + ((leaf * B_SZ + b) * CHUNKS + chunk)] = s;
  }
}

// ---------------------------------------------------------------------------
// Kernel 3: psi0/psi1 from chunk partials -> (sign, logabs).
// Output-layer bias c contributes c * sum_p sign(p) == 0 for n >= 2 (dropped).
// ---------------------------------------------------------------------------
__global__ void finalize_kernel(const float* __restrict__ ws, float* __restrict__ out) {
  int b = threadIdx.x;
  if (b >= B_SZ) return;
  const float* part = ws + WS_PART;
  float psi0 = 0.f, psi1 = 0.f;
#pragma unroll
  for (int c = 0; c < CHUNKS; ++c) {
    psi0 += part[b * CHUNKS + c];
    psi1 += part[(B_SZ + b) * CHUNKS + c];
  }
  float s0 = (psi0 > 0.f) ? 1.f : ((psi0 < 0.f) ? -1.f : 0.f);
  float s1 = (psi1 > 0.f) ? 1.f : ((psi1 < 0.f) ? -1.f : 0.f);
  out[b]        = s0 * s1;
  out[B_SZ + b] = logf(fabsf(psi0)) + logf(fabsf(psi1));
}

// ---------------------------------------------------------------------------
extern "C" void kernel_launch(void* const* d_in, const int* in_sizes, int n_in,
                              void* d_out, int out_size, void* d_ws, size_t ws_size,
                              hipStream_t stream) {
  (void)in_sizes; (void)n_in; (void)out_size; (void)ws_size;
  const float* x0 = (const float*)d_in[0];
  const float* x1 = (const float*)d_in[1];
  const float* W0 = (const float*)d_in[2];
  const float* b0 = (const float*)d_in[3];
  const float* V0 = (const float*)d_in[4];
  /* c0 = d_in[5]: contributes c * sum(signs) == 0, unused */
  const float* W1 = (const float*)d_in[6];
  const float* b1 = (const float*)d_in[7];
  const float* V1 = (const float*)d_in[8];
  /* c1 = d_in[9]: unused (zero contribution) */

  float* ws  = (float*)d_ws;
  float* out = (float*)d_out;

  build_perm_codes<<<dim3((NPERM + 255) / 256), dim3(256), 0, stream>>>((uint32_t*)d_ws);
  precompute_P_wmma<<<dim3(56, 7, 2), dim3(256), 0, stream>>>(x0, W0, x1, W1, ws);
  perm_sum_kernel<<<dim3(CHUNKS, B_SZ, 2), dim3(256), 0, stream>>>(b0, V0, b1, V1, ws);
  finalize_kernel<<<dim3(1), dim3(128), 0, stream>>>(ws, out);
}